// RateAdaptionEncoder_20547123544240
// MI455X (gfx1250) — compile-verified
//
#include <hip/hip_runtime.h>

typedef __attribute__((ext_vector_type(2))) float v2f;
typedef __attribute__((ext_vector_type(8))) float v8f;

#define NTOK 4096      // B*H*W tokens
#define CCH  256       // input channels (K)
#define RR   256       // max rate (N)
#define NEXP 8         // number of experts
#define HW   1024      // H*W
#define TILE 32        // tokens per GEMM tile (2 x WMMA M)
#define MAXTILES 128   // NTOK / TILE
#define LPITCH 260     // LDS row pitch in floats (256 + 4 -> conflict-free)

// ---------------- pass 0: zero per-expert counters ----------------
__global__ void zero_counts(int* __restrict__ cnt) {
    if (threadIdx.x < NEXP) cnt[threadIdx.x] = 0;
}

// ---------------- pass 1: bin tokens by expert --------------------
__global__ void bin_tokens(const int* __restrict__ idx,
                           int* __restrict__ cnt,
                           int* __restrict__ list) {
    int t = blockIdx.x * blockDim.x + threadIdx.x;
    if (t < NTOK) {
        int e = idx[t];
        int slot = atomicAdd(&cnt[e], 1);
        list[e * NTOK + slot] = t;
    }
}

// ---------------- pass 2: grouped GEMM via WMMA fp32 --------------
// grid: NEXP * 128 blocks; block b -> expert e = b>>7, tile = b&127
// block: 128 threads = 4 waves; wave w owns cols [w*64, w*64+64), rows 0..31
// Each wave: 2 M-fragments x 4 N-tiles = 8 WMMA accumulators; each B fragment
// feeds 2 WMMAs (B-reuse across the two M fragments -> half the VMEM loads).
__global__ __launch_bounds__(128)
void rate_adaption_gemm(const float* __restrict__ x,       // (B, C, H, W)
                        const float* __restrict__ weight,  // (NEXP, C, R)
                        const float* __restrict__ bias,    // (NEXP, R)
                        const int*   __restrict__ rate_choice, // (NEXP,)
                        const int*   __restrict__ cnt,
                        const int*   __restrict__ list,
                        float* __restrict__ out) {         // [x_masked | mask]
    const int e    = blockIdx.x >> 7;
    const int tile = blockIdx.x & (MAXTILES - 1);
    const int n    = cnt[e];
    const int start = tile * TILE;
    if (start >= n) return;                 // block-uniform: EXEC all-1s below
    const int nvalid = min(TILE, n - start);

    __shared__ float xs[TILE * LPITCH];
    __shared__ int   stok[TILE];

    const int tid = threadIdx.x;
    if (tid < TILE)
        stok[tid] = (tid < nvalid) ? list[e * NTOK + start + tid] : -1;
    __syncthreads();

    // Stage gathered x rows (token-major, channel transpose-gather) into LDS.
    for (int i = tid; i < TILE * CCH; i += 128) {
        int m = i >> 8;            // token row in tile
        int c = i & 255;           // channel
        int t = stok[m];
        float v = 0.0f;
        if (t >= 0)
            v = x[(t >> 10) * (CCH * HW) + c * HW + (t & (HW - 1))];
        xs[m * LPITCH + c] = v;
    }
    __syncthreads();

    const int lane  = tid & 31;
    const int wv    = tid >> 5;            // wave id 0..3
    const int nlo   = lane & 15;           // N (col) / M (row) sub-index
    const int khalf = (lane >> 4) << 1;    // K offset within 4-wide step: 0 or 2

    const float* wb = weight + e * (CCH * RR);
    const int col0 = wv * 64 + nlo;

    v8f acc[2][4] = {};

    for (int k0 = 0; k0 < CCH; k0 += 4) {
        // Two A fragments: rows 0..15 and 16..31 of the token tile (from LDS)
        v2f a0, a1;
        a0.x = xs[nlo * LPITCH + k0 + khalf];
        a0.y = xs[nlo * LPITCH + k0 + khalf + 1];
        a1.x = xs[(nlo + 16) * LPITCH + k0 + khalf];
        a1.y = xs[(nlo + 16) * LPITCH + k0 + khalf + 1];
        const float* wr0 = wb + (k0 + khalf) * RR + col0;   // row k0+khalf
#pragma unroll
        for (int ct = 0; ct < 4; ++ct) {
            // B fragment: 4x16 fp32 tile of expert weight (L2-resident),
            // reused by both M fragments.
            v2f b;
            b.x = wr0[ct * 16];
            b.y = wr0[ct * 16 + RR];
            acc[0][ct] = __builtin_amdgcn_wmma_f32_16x16x4_f32(
                false, a0, false, b, (short)0, acc[0][ct], false, false);
            acc[1][ct] = __builtin_amdgcn_wmma_f32_16x16x4_f32(
                false, a1, false, b, (short)0, acc[1][ct], false, false);
        }
    }

    // Epilogue: bias + rate mask, scatter to (B, R, H, W); also write mask plane.
    const int rc = rate_choice[e];
    const int mbase = (lane >> 4) << 3;    // C/D layout: lanes 16-31 hold M = v+8
    float* mout = out + NTOK * RR;

#pragma unroll
    for (int f = 0; f < 2; ++f) {
#pragma unroll
        for (int ct = 0; ct < 4; ++ct) {
            const int r = wv * 64 + ct * 16 + nlo;
            const float bval = bias[e * RR + r];
            const float keep = (r < rc) ? 1.0f : 0.0f;
#pragma unroll
            for (int v = 0; v < 8; ++v) {
                const int m = f * 16 + mbase + v;
                if (m < nvalid) {
                    const int t = stok[m];
                    const int oi = (t >> 10) * (RR * HW) + r * HW + (t & (HW - 1));
                    out[oi]  = (acc[f][ct][v] + bval) * keep;
                    mout[oi] = keep;
                }
            }
        }
    }
}

extern "C" void kernel_launch(void* const* d_in, const int* in_sizes, int n_in,
                              void* d_out, int out_size, void* d_ws, size_t ws_size,
                              hipStream_t stream) {
    const float* x           = (const float*)d_in[0];
    const int*   indexes     = (const int*)  d_in[1];
    const float* weight      = (const float*)d_in[2];
    const float* bias        = (const float*)d_in[3];
    const int*   rate_choice = (const int*)  d_in[4];

    int* cnt  = (int*)d_ws;          // 8 counters
    int* list = cnt + NEXP;          // 8 * 4096 token ids

    zero_counts<<<1, 32, 0, stream>>>(cnt);
    bin_tokens<<<NTOK / 256, 256, 0, stream>>>(indexes, cnt, list);
    rate_adaption_gemm<<<NEXP * MAXTILES, 128, 0, stream>>>(
        x, weight, bias, rate_choice, cnt, list, (float*)d_out);
}